// GINNet_15015205667106
// MI455X (gfx1250) — compile-verified
//
#include <hip/hip_runtime.h>
#include <hip/hip_bf16.h>

#define N_NODES  50000
#define N_EDGES  800000
#define NODE_DIM 128
#define IN_CH    64
#define OUT_CH   16
#define N_GRAPHS 512
#define N_LAYERS 5
#define BN_EPS   1e-5f

typedef __attribute__((ext_vector_type(2))) float v2f;
typedef __attribute__((ext_vector_type(8))) float v8f;

__device__ __forceinline__ v8f wmma_f32(v2f a, v2f b, v8f c) {
    // D = A(16x4,f32) * B(4x16,f32) + C(16x16,f32)
    return __builtin_amdgcn_wmma_f32_16x16x4_f32(
        /*neg_a=*/false, a, /*neg_b=*/false, b,
        /*c_mod=*/(short)0, c, /*reuse_a=*/false, /*reuse_b=*/false);
}

// ---------------------------------------------------------------------------
// Encoder: H[N,64] = X[N,128] @ W[128,64] + b
// block = 128 threads (4 waves), 64 rows per block, W staged in LDS.
// ---------------------------------------------------------------------------
__global__ __launch_bounds__(128) void enc_gemm(
        const float* __restrict__ X, const float* __restrict__ W,
        const float* __restrict__ bias, float* __restrict__ H) {
    __shared__ float sW[NODE_DIM * IN_CH];          // 32 KB
    const int tid = threadIdx.x;
    for (int i = tid; i < NODE_DIM * IN_CH; i += 128) sW[i] = W[i];
    __syncthreads();

    const int wave = tid >> 5, lane = tid & 31;
    const int half = lane >> 4, l16 = lane & 15;
    const int rowBase = blockIdx.x * 64 + wave * 16;

    int arow = rowBase + l16;
    if (arow > N_NODES - 1) arow = N_NODES - 1;     // clamped load, masked store
    const float* __restrict__ Arow = X + (long)arow * NODE_DIM;

    v8f acc[4] = {};
    for (int k = 0; k < NODE_DIM; k += 4) {
        const int kk = k + 2 * half;
        v2f a;
        a.x = Arow[kk];
        a.y = Arow[kk + 1];
#pragma unroll
        for (int n = 0; n < 4; ++n) {
            v2f b;
            b.x = sW[kk * IN_CH + n * 16 + l16];
            b.y = sW[(kk + 1) * IN_CH + n * 16 + l16];
            acc[n] = wmma_f32(a, b, acc[n]);
        }
    }
#pragma unroll
    for (int n = 0; n < 4; ++n) {
        const float bv = bias[n * 16 + l16];
#pragma unroll
        for (int v = 0; v < 8; ++v) {
            const int r = rowBase + v + 8 * half;
            if (r < N_NODES) H[(long)r * IN_CH + n * 16 + l16] = acc[n][v] + bv;
        }
    }
}

// ---------------------------------------------------------------------------
// Edge aggregation: AGG[dst] += H[src].  16 lanes per edge, float4 per lane.
// ---------------------------------------------------------------------------
__global__ void edge_scatter(const int* __restrict__ src, const int* __restrict__ dst,
                             const float* __restrict__ H, float* __restrict__ AGG,
                             int nEdges) {
    const long t = (long)blockIdx.x * blockDim.x + threadIdx.x;
    const long e = t >> 4;
    const int  q = (int)(t & 15);
    if (e >= nEdges) return;
    const int s = src[e], d = dst[e];
    const float4 val = *(const float4*)(H + (long)s * IN_CH + q * 4);
    float* out = AGG + (long)d * IN_CH + q * 4;
    atomicAdd(out + 0, val.x);
    atomicAdd(out + 1, val.y);
    atomicAdd(out + 2, val.z);
    atomicAdd(out + 3, val.w);
}

// ---------------------------------------------------------------------------
// Fused GIN MLP: Z = relu( relu((H+AGG)@W1 + b1) @ W2 + b2 )
// Intermediate tile lives in LDS; BN sum/sumsq reduced from accumulators.
// ---------------------------------------------------------------------------
__global__ __launch_bounds__(128) void gin_mlp(
        const float* __restrict__ H, const float* __restrict__ AGG,
        const float* __restrict__ W1, const float* __restrict__ b1,
        const float* __restrict__ W2, const float* __restrict__ b2,
        float* __restrict__ Z, float* __restrict__ tsum, float* __restrict__ tsumsq) {
    __shared__ float sW1[IN_CH * IN_CH];            // 16 KB
    __shared__ float sW2[IN_CH * IN_CH];            // 16 KB
    __shared__ float sT[64 * IN_CH];                // 16 KB intermediate tile
    const int tid = threadIdx.x;
    for (int i = tid; i < IN_CH * IN_CH; i += 128) { sW1[i] = W1[i]; sW2[i] = W2[i]; }
    __syncthreads();

    const int wave = tid >> 5, lane = tid & 31;
    const int half = lane >> 4, l16 = lane & 15;
    const int rowBase = blockIdx.x * 64 + wave * 16;

    int arow = rowBase + l16;
    if (arow > N_NODES - 1) arow = N_NODES - 1;
    const float* __restrict__ hrow = H + (long)arow * IN_CH;
    const float* __restrict__ grow = AGG + (long)arow * IN_CH;

    // GEMM1: t = relu((h+agg) @ W1 + b1)
    v8f acc[4] = {};
    for (int k = 0; k < IN_CH; k += 4) {
        const int kk = k + 2 * half;
        v2f a;
        a.x = hrow[kk] + grow[kk];
        a.y = hrow[kk + 1] + grow[kk + 1];
#pragma unroll
        for (int n = 0; n < 4; ++n) {
            v2f b;
            b.x = sW1[kk * IN_CH + n * 16 + l16];
            b.y = sW1[(kk + 1) * IN_CH + n * 16 + l16];
            acc[n] = wmma_f32(a, b, acc[n]);
        }
    }
    const int trowBase = wave * 16;
#pragma unroll
    for (int n = 0; n < 4; ++n) {
        const float bv = b1[n * 16 + l16];
#pragma unroll
        for (int v = 0; v < 8; ++v) {
            float val = acc[n][v] + bv;
            val = val > 0.f ? val : 0.f;
            sT[(trowBase + v + 8 * half) * IN_CH + n * 16 + l16] = val;
        }
    }
    __syncthreads();

    // GEMM2: z = relu(t @ W2 + b2), plus BN sum/sumsq per channel
    v8f acc2[4] = {};
    const float* __restrict__ trow = sT + (trowBase + l16) * IN_CH;
    for (int k = 0; k < IN_CH; k += 4) {
        const int kk = k + 2 * half;
        v2f a;
        a.x = trow[kk];
        a.y = trow[kk + 1];
#pragma unroll
        for (int n = 0; n < 4; ++n) {
            v2f b;
            b.x = sW2[kk * IN_CH + n * 16 + l16];
            b.y = sW2[(kk + 1) * IN_CH + n * 16 + l16];
            acc2[n] = wmma_f32(a, b, acc2[n]);
        }
    }
#pragma unroll
    for (int n = 0; n < 4; ++n) {
        const float bv = b2[n * 16 + l16];
        float csum = 0.f, csq = 0.f;
#pragma unroll
        for (int v = 0; v < 8; ++v) {
            const int r = rowBase + v + 8 * half;
            float val = acc2[n][v] + bv;
            val = val > 0.f ? val : 0.f;
            if (r < N_NODES) {
                Z[(long)r * IN_CH + n * 16 + l16] = val;
                csum += val;
                csq  += val * val;
            }
        }
        // lanes L and L+16 cover the same column -> combine halves, one atomic
        csum += __shfl_xor(csum, 16, 32);
        csq  += __shfl_xor(csq,  16, 32);
        if (half == 0) {
            atomicAdd(&tsum[n * 16 + l16], csum);
            atomicAdd(&tsumsq[n * 16 + l16], csq);
        }
    }
}

// ---------------------------------------------------------------------------
// BN finalize (1 block, 64 threads): scale = g*rsqrt(var+eps); shift = b - mean*scale
// ---------------------------------------------------------------------------
__global__ void bn_finalize(const float* __restrict__ tsum, const float* __restrict__ tsumsq,
                            const float* __restrict__ gamma, const float* __restrict__ beta,
                            float* __restrict__ scale, float* __restrict__ shift) {
    const int c = threadIdx.x;
    const float inv_n = 1.f / (float)N_NODES;
    const float mean = tsum[c] * inv_n;
    const float var  = tsumsq[c] * inv_n - mean * mean;
    const float sc   = gamma[c] * rsqrtf(var + BN_EPS);
    scale[c] = sc;
    shift[c] = beta[c] - mean * sc;
}

__global__ void bn_apply(const float* __restrict__ Z, const float* __restrict__ scale,
                         const float* __restrict__ shift, float* __restrict__ H, int n) {
    const int i = blockIdx.x * blockDim.x + threadIdx.x;
    if (i < n) {
        const int c = i & (IN_CH - 1);
        H[i] = Z[i] * scale[c] + shift[c];
    }
}

// ---------------------------------------------------------------------------
// Global add pool: G[batch[n]] += H[n].  16 lanes/node, float4 per lane.
// ---------------------------------------------------------------------------
__global__ void pool_scatter(const int* __restrict__ batch, const float* __restrict__ H,
                             float* __restrict__ G) {
    const long t = (long)blockIdx.x * blockDim.x + threadIdx.x;
    const long nidx = t >> 4;
    const int  q = (int)(t & 15);
    if (nidx >= N_NODES) return;
    const int g = batch[nidx];
    const float4 val = *(const float4*)(H + nidx * IN_CH + q * 4);
    float* out = G + (long)g * IN_CH + q * 4;
    atomicAdd(out + 0, val.x);
    atomicAdd(out + 1, val.y);
    atomicAdd(out + 2, val.z);
    atomicAdd(out + 3, val.w);
}

// ---------------------------------------------------------------------------
// Head: OUT[512,16] = relu(G @ fc1 + b1) @ fc2 + b2.  8 blocks x 64 rows.
// ---------------------------------------------------------------------------
__global__ __launch_bounds__(128) void head(
        const float* __restrict__ G, const float* __restrict__ fc1W,
        const float* __restrict__ fc1b, const float* __restrict__ fc2W,
        const float* __restrict__ fc2b, float* __restrict__ OUT) {
    __shared__ float sW1[IN_CH * IN_CH];
    __shared__ float sW2[IN_CH * OUT_CH];
    __shared__ float sT[64 * IN_CH];
    const int tid = threadIdx.x;
    for (int i = tid; i < IN_CH * IN_CH; i += 128) sW1[i] = fc1W[i];
    for (int i = tid; i < IN_CH * OUT_CH; i += 128) sW2[i] = fc2W[i];
    __syncthreads();

    const int wave = tid >> 5, lane = tid & 31;
    const int half = lane >> 4, l16 = lane & 15;
    const int rowBase = blockIdx.x * 64 + wave * 16;   // always < 512

    const float* __restrict__ grow = G + (long)(rowBase + l16) * IN_CH;
    v8f acc[4] = {};
    for (int k = 0; k < IN_CH; k += 4) {
        const int kk = k + 2 * half;
        v2f a;
        a.x = grow[kk];
        a.y = grow[kk + 1];
#pragma unroll
        for (int n = 0; n < 4; ++n) {
            v2f b;
            b.x = sW1[kk * IN_CH + n * 16 + l16];
            b.y = sW1[(kk + 1) * IN_CH + n * 16 + l16];
            acc[n] = wmma_f32(a, b, acc[n]);
        }
    }
    const int trowBase = wave * 16;
#pragma unroll
    for (int n = 0; n < 4; ++n) {
        const float bv = fc1b[n * 16 + l16];
#pragma unroll
        for (int v = 0; v < 8; ++v) {
            float val = acc[n][v] + bv;
            val = val > 0.f ? val : 0.f;
            sT[(trowBase + v + 8 * half) * IN_CH + n * 16 + l16] = val;
        }
    }
    __syncthreads();

    v8f acc2 = {};
    const float* __restrict__ trow = sT + (trowBase + l16) * IN_CH;
    for (int k = 0; k < IN_CH; k += 4) {
        const int kk = k + 2 * half;
        v2f a;
        a.x = trow[kk];
        a.y = trow[kk + 1];
        v2f b;
        b.x = sW2[kk * OUT_CH + l16];
        b.y = sW2[(kk + 1) * OUT_CH + l16];
        acc2 = wmma_f32(a, b, acc2);
    }
    const float bv = fc2b[l16];
#pragma unroll
    for (int v = 0; v < 8; ++v) {
        const int r = rowBase + v + 8 * half;
        OUT[(long)r * OUT_CH + l16] = acc2[v] + bv;
    }
}

// ---------------------------------------------------------------------------
extern "C" void kernel_launch(void* const* d_in, const int* in_sizes, int n_in,
                              void* d_out, int out_size, void* d_ws, size_t ws_size,
                              hipStream_t stream) {
    (void)in_sizes; (void)n_in; (void)out_size; (void)ws_size;

    const float* x        = (const float*)d_in[0];
    const int*   edge_idx = (const int*)d_in[1];
    const int*   batch    = (const int*)d_in[2];
    const float* enc_W    = (const float*)d_in[3];
    const float* enc_b    = (const float*)d_in[4];
    const float* conv_W1  = (const float*)d_in[5];
    const float* conv_b1  = (const float*)d_in[6];
    const float* conv_W2  = (const float*)d_in[7];
    const float* conv_b2  = (const float*)d_in[8];
    const float* bn_gamma = (const float*)d_in[9];
    const float* bn_beta  = (const float*)d_in[10];
    const float* fc1_W    = (const float*)d_in[11];
    const float* fc1_b    = (const float*)d_in[12];
    const float* fc2_W    = (const float*)d_in[13];
    const float* fc2_b    = (const float*)d_in[14];
    float* out = (float*)d_out;

    const int* src = edge_idx;             // edge_index[0,:]
    const int* dst = edge_idx + N_EDGES;   // edge_index[1,:]

    // Workspace carve-up
    char* ws = (char*)d_ws;
    const size_t featBytes = (size_t)N_NODES * IN_CH * sizeof(float);
    float* H   = (float*)ws; ws += featBytes;
    float* Z   = (float*)ws; ws += featBytes;
    float* AGG = (float*)ws; ws += featBytes;
    float* G      = (float*)ws; ws += (size_t)N_GRAPHS * IN_CH * sizeof(float);
    float* tsum   = (float*)ws; ws += IN_CH * sizeof(float);
    float* tsumsq = (float*)ws; ws += IN_CH * sizeof(float);
    float* scale  = (float*)ws; ws += IN_CH * sizeof(float);
    float* shift  = (float*)ws; ws += IN_CH * sizeof(float);

    const int nodeBlocks = (N_NODES + 63) / 64;

    enc_gemm<<<nodeBlocks, 128, 0, stream>>>(x, enc_W, enc_b, H);

    for (int l = 0; l < N_LAYERS; ++l) {
        hipMemsetAsync(AGG, 0, featBytes, stream);
        hipMemsetAsync(tsum, 0, IN_CH * sizeof(float), stream);
        hipMemsetAsync(tsumsq, 0, IN_CH * sizeof(float), stream);

        edge_scatter<<<(N_EDGES * 16 + 255) / 256, 256, 0, stream>>>(src, dst, H, AGG, N_EDGES);

        gin_mlp<<<nodeBlocks, 128, 0, stream>>>(
            H, AGG,
            conv_W1 + (size_t)l * IN_CH * IN_CH, conv_b1 + (size_t)l * IN_CH,
            conv_W2 + (size_t)l * IN_CH * IN_CH, conv_b2 + (size_t)l * IN_CH,
            Z, tsum, tsumsq);

        bn_finalize<<<1, IN_CH, 0, stream>>>(tsum, tsumsq,
            bn_gamma + (size_t)l * IN_CH, bn_beta + (size_t)l * IN_CH, scale, shift);

        bn_apply<<<(N_NODES * IN_CH + 255) / 256, 256, 0, stream>>>(Z, scale, shift, H,
                                                                    N_NODES * IN_CH);
    }

    hipMemsetAsync(G, 0, (size_t)N_GRAPHS * IN_CH * sizeof(float), stream);
    pool_scatter<<<(N_NODES * 16 + 255) / 256, 256, 0, stream>>>(batch, H, G);
    head<<<N_GRAPHS / 64, 128, 0, stream>>>(G, fc1_W, fc1_b, fc2_W, fc2_b, out);
}